// ChamferLossV2_20134806683973
// MI455X (gfx1250) — compile-verified
//
#include <hip/hip_runtime.h>
#include <hip/hip_bf16.h>

typedef __attribute__((ext_vector_type(2))) float v2f;
typedef __attribute__((ext_vector_type(4))) float v4f;
typedef __attribute__((ext_vector_type(8))) float v8f;

#define NP      256   // particles per (bs,h)
#define OD      10    // OBS_DIM
#define FD      4     // feature dims (obs[...,5:9])
#define WAVES   8
#define THREADS 256

__global__ void chamfer_init_out(float* out) {
    if (threadIdx.x < 2) out[threadIdx.x] = 0.0f;
}

// Order-preserving map from float bits to uint32 (total order incl. negatives)
__device__ __forceinline__ unsigned orderKey(float f) {
    unsigned u = __float_as_uint(f);
    return u ^ (unsigned)(((int)u >> 31) | 0x80000000);
}

// lane l <-> lane l^16 exchange of a 64-bit value
__device__ __forceinline__ unsigned long long xor16_u64(unsigned long long x) {
#if __has_builtin(__builtin_amdgcn_permlanex16)
    int lo = (int)(unsigned)x;
    int hi = (int)(unsigned)(x >> 32);
    lo = __builtin_amdgcn_permlanex16(lo, lo, 0x76543210, 0xfedcba98, true, false);
    hi = __builtin_amdgcn_permlanex16(hi, hi, 0x76543210, 0xfedcba98, true, false);
    return ((unsigned long long)(unsigned)hi << 32) | (unsigned)lo;
#else
    return __shfl_xor(x, 16, 32);
#endif
}

__global__ __launch_bounds__(THREADS)
void chamfer_loss_kernel(const float* __restrict__ preds,
                         const float* __restrict__ targ,
                         float* __restrict__ out) {
    __shared__ float sObsP[NP * OD];
    __shared__ float sObsT[NP * OD];
    __shared__ float sFp[NP * FD];
    __shared__ float sFt[NP * FD];
    __shared__ float sNp[NP];
    __shared__ float sNt[NP];
    __shared__ unsigned long long rowKey[NP];  // argmin_j P[i][j]  -> idx2
    __shared__ unsigned long long colKey[NP];  // argmin_i P[i][j]  -> idx1
    __shared__ float sred[THREADS];

    const int b = blockIdx.x;        // b = bs_idx*32 + h, 512 blocks
    const int t = threadIdx.x;
    const long long rowbase = (long long)b * 2564;

    // ---- stage obs rows, features, norms into LDS ----
    {
        const float* op = preds + rowbase + 4 + (long long)t * OD;
        const float* ot = targ  + rowbase + 4 + (long long)t * OD;
        float np_ = 0.0f, nt_ = 0.0f;
#pragma unroll
        for (int k = 0; k < OD; ++k) {
            float vp = op[k];
            float vt = ot[k];
            sObsP[t * OD + k] = vp;
            sObsT[t * OD + k] = vt;
            if (k >= 5 && k < 9) {
                sFp[t * FD + (k - 5)] = vp;
                sFt[t * FD + (k - 5)] = vt;
                np_ += vp * vp;
                nt_ += vt * vt;
            }
        }
        sNp[t] = np_;
        sNt[t] = nt_;
        colKey[t] = ~0ULL;
    }
    __syncthreads();

    const int wave = t >> 5;
    const int lane = t & 31;
    const int l    = lane & 15;
    const int half = lane >> 4;

    // ---- pairwise argmins via WMMA f32 16x16x4 ----
    // P[i][j] = |fi|^2 + |gj|^2 - 2 fi.gj ; drop the term constant in the argmin axis.
#pragma unroll
    for (int s = 0; s < 2; ++s) {
        const int ti = wave + 8 * s;          // this wave exclusively owns tile-row ti
        const int arow = ti * 16 + l;

        // A operand, strip-invariant (lanes 0-15: K=0,1 ; lanes 16-31: K=2,3)
        v2f a;
        a[0] = sFp[arow * FD + 2 * half + 0];
        a[1] = sFp[arow * FD + 2 * half + 1];

        // |fi|^2 for the 8 C-rows of this half, strip-invariant (two b128 loads)
        const v4f npA = *(const v4f*)&sNp[ti * 16 + 8 * half + 0];
        const v4f npB = *(const v4f*)&sNp[ti * 16 + 8 * half + 4];
        float npi[8] = {npA[0], npA[1], npA[2], npA[3], npB[0], npB[1], npB[2], npB[3]};

        unsigned long long rk[8];
#pragma unroll
        for (int v = 0; v < 8; ++v) rk[v] = ~0ULL;

        for (int tj = 0; tj < 16; ++tj) {     // uniform loop: EXEC all-1s at WMMA
            const int bcol = tj * 16 + l;
            v2f bm;
            bm[0] = sFt[bcol * FD + 2 * half + 0];
            bm[1] = sFt[bcol * FD + 2 * half + 1];
            const float ntj = sNt[bcol];

            v8f c = {0.f, 0.f, 0.f, 0.f, 0.f, 0.f, 0.f, 0.f};
#if __has_builtin(__builtin_amdgcn_wmma_f32_16x16x4_f32)
            c = __builtin_amdgcn_wmma_f32_16x16x4_f32(false, a, false, bm,
                                                      (short)0, c, false, false);
#else
#pragma unroll
            for (int v = 0; v < 8; ++v) {
                const int ir = ti * 16 + 8 * half + v;
                float d = 0.0f;
#pragma unroll
                for (int k = 0; k < FD; ++k) d += sFp[ir * FD + k] * sFt[bcol * FD + k];
                c[v] = d;
            }
#endif

            unsigned long long ck = ~0ULL;    // running col-min key for column bcol
#pragma unroll
            for (int v = 0; v < 8; ++v) {
                const float dot = c[v];
                // row argmin over j: value ~ nt_j - 2*dot
                const unsigned long long krk =
                    ((unsigned long long)orderKey(fmaf(-2.0f, dot, ntj)) << 32) |
                    (unsigned)bcol;
                if (krk < rk[v]) rk[v] = krk;             // v_min_u64
                // col argmin over i: value ~ np_i - 2*dot
                const int irow = ti * 16 + 8 * half + v;
                const unsigned long long kck =
                    ((unsigned long long)orderKey(fmaf(-2.0f, dot, npi[v])) << 32) |
                    (unsigned)irow;
                if (kck < ck) ck = kck;                   // v_min_u64
            }
            // fold the two halves (same column), then one ds_min_u64 per tile.
            // Both halves carry identical (addr,val): no EXEC divergence needed.
            {
                unsigned long long o = xor16_u64(ck);
                if (o < ck) ck = o;
                atomicMin(&colKey[bcol], ck);
            }
        }

        // one butterfly per row per strip; rows are wave-exclusive -> plain store
#pragma unroll
        for (int v = 0; v < 8; ++v) {
            unsigned long long k0 = rk[v];
#pragma unroll
            for (int m = 1; m < 16; m <<= 1) {
                unsigned long long o = __shfl_xor(k0, m, 16);
                if (o < k0) k0 = o;
            }
            if (l == 0) rowKey[ti * 16 + 8 * half + v] = k0;
        }
    }
    __syncthreads();

    // ---- gather + weighted L1 accumulation ----
    float local;
    {
        const int i2 = (int)(rowKey[t] & 0xFFFFFFFFull);  // idx2[t]: nearest targ for pred t
        const int i1 = (int)(colKey[t] & 0xFFFFFFFFull);  // idx1[t]: nearest pred for targ t
        float s1 = 0.0f, s2 = 0.0f;
#pragma unroll
        for (int k = 0; k < OD; ++k) {
            s1 += fabsf(sObsP[i1 * OD + k] - sObsT[t * OD + k]);  // particle_dist1
            s2 += fabsf(sObsT[i2 * OD + k] - sObsP[t * OD + k]);  // particle_dist2
        }
        local = 3.0f * s1 + s2;   // TARGET_WEIGHT*pd1 + pd2
    }
    sred[t] = local;
    __syncthreads();
#pragma unroll
    for (int s = THREADS / 2; s > 0; s >>= 1) {
        if (t < s) sred[t] += sred[t + s];
        __syncthreads();
    }

    if (t == 0) {
        // chamfer.mean(): /(TARGET_WEIGHT+1)=4 and /(512*256*10) elements
        float contrib = sred[0] * (0.25f / 1310720.0f);

        // action terms for this (bs,h)
        const float* ap = preds + rowbase;
        const float* at = targ + rowbase;
        float al1 = 0.0f;
#pragma unroll
        for (int k = 0; k < 4; ++k) al1 += fabsf(ap[k] - at[k]);
        al1 *= 0.25f;                              // mean over ACTION_DIM
        const int h = b & 31;
        const float w = (h == 1) ? 10.0f : 1.0f;   // A0_WEIGHT at horizon index 1
        contrib += al1 * w * (1.0f / 512.0f);      // action_loss mean over (16,32)
        atomicAdd(&out[0], contrib);
        if (h == 1) atomicAdd(&out[1], al1 * (1.0f / 16.0f));  // a0_loss (pre-weight)
    }
}

extern "C" void kernel_launch(void* const* d_in, const int* in_sizes, int n_in,
                              void* d_out, int out_size, void* d_ws, size_t ws_size,
                              hipStream_t stream) {
    const float* preds = (const float*)d_in[0];
    const float* targ  = (const float*)d_in[1];
    float* out = (float*)d_out;

    hipLaunchKernelGGL(chamfer_init_out, dim3(1), dim3(32), 0, stream, out);
    hipLaunchKernelGGL(chamfer_loss_kernel, dim3(512), dim3(THREADS), 0, stream,
                       preds, targ, out);
}